// STHGCN_21887153340795
// MI455X (gfx1250) — compile-verified
//
#include <hip/hip_runtime.h>
#include <hip/hip_bf16.h>

typedef float v2f __attribute__((ext_vector_type(2)));
typedef float v8f __attribute__((ext_vector_type(8)));

#define N_NODES 512
#define EDGES   128
#define BATCH   32
#define SEQ     32
#define DIN     16
#define HD      64
#define ROWS    (BATCH * N_NODES)   // 16384
#define NH      (N_NODES * HD)      // 32768 per batch
#define GSTRIDE 516                 // padded LDS stride (4 mod 64 -> conflict-free 16x4 frag reads)

// ---------------- graph operator G = (Hn * de^-1) @ Hn^T ----------------

__global__ void k_deg(const float* __restrict__ H, float* dv, float* de) {
  int tid = threadIdx.x;                       // 512 threads
  float s = 0.f;
  for (int e = 0; e < EDGES; ++e) s += H[tid * EDGES + e];
  dv[tid] = s;
  if (tid < EDGES) {
    float t = 0.f;
    for (int n = 0; n < N_NODES; ++n) t += H[n * EDGES + tid];
    de[tid] = t;
  }
}

__global__ __launch_bounds__(512) void k_buildG(const float* __restrict__ H,
                                                const float* __restrict__ dv,
                                                const float* __restrict__ de,
                                                float* __restrict__ G) {
  int n = blockIdx.x, m = threadIdx.x;
  float s = 0.f;
  for (int e = 0; e < EDGES; ++e) s += H[n * EDGES + e] * H[m * EDGES + e] / de[e];
  G[n * N_NODES + m] = s * rsqrtf(dv[n]) * rsqrtf(dv[m]);
}

__global__ void k_fill(float* p, float v, int n) {
  int i = blockIdx.x * blockDim.x + threadIdx.x;
  if (i < n) p[i] = v;
}

// ---------------- GEMM: concat(x_t, h) @ W  via exact-f32 WMMA ----------------
// out (ROWS x C) = [x_t | h] (ROWS x 80) @ W (80 x C), C in {128, 64}

__global__ __launch_bounds__(256) void k_catmm(const float* __restrict__ inputs, int t,
                                               const float* __restrict__ h,
                                               const float* __restrict__ W, int C,
                                               float* __restrict__ out) {
  int wave = (blockIdx.x * blockDim.x + threadIdx.x) >> 5;
  int lane = threadIdx.x & 31;
  int tiles_c = C >> 4;
  int tm = wave / tiles_c, tc = wave % tiles_c;
  int r16 = lane & 15, hi = lane >> 4;
  int m = tm * 16 + r16;                       // A-matrix row for this lane
  int b = m >> 9, n = m & 511;
  const float* xt = inputs + (size_t)b * (SEQ * N_NODES * DIN)
                           + (size_t)t * (N_NODES * DIN) + (size_t)n * DIN;
  const float* hb = h + (size_t)b * NH + (size_t)n * HD;
  int col = tc * 16 + r16;                     // B-matrix col for this lane
  v8f acc = {};
  for (int k0 = 0; k0 < DIN + HD; k0 += 4) {
    int ka = k0 + 2 * hi;                      // lane holds A[m][ka], A[m][ka+1]
    float a0 = (ka     < DIN) ? xt[ka]          : hb[ka - DIN];
    float a1 = (ka + 1 < DIN) ? xt[ka + 1]      : hb[ka + 1 - DIN];
    v2f av = {a0, a1};
    v2f bv = {W[ka * C + col], W[(ka + 1) * C + col]};
    acc = __builtin_amdgcn_wmma_f32_16x16x4_f32(false, av, false, bv, (short)0, acc,
                                                false, false);
  }
  float* o = out + (size_t)(tm * 16 + 8 * hi) * C + col;
#pragma unroll
  for (int r = 0; r < 8; ++r) o[(size_t)r * C] = acc[r];
}

// ---------------- GEMM: out[b] = G (512x512) @ x[b] (512xC), LDS-staged G ----------------
// grid.x = 32 row tiles of G; grid.y * 8 waves = BATCH * (C/16) column tiles

__global__ __launch_bounds__(256) void k_gconv(const float* __restrict__ G,
                                               const float* __restrict__ x, int C,
                                               float* __restrict__ out) {
  __shared__ float Gs[16 * GSTRIDE];
  int ntile = blockIdx.x;
  for (int i = threadIdx.x; i < 16 * N_NODES; i += blockDim.x) {
    int rr = i >> 9, kk = i & 511;
    Gs[rr * GSTRIDE + kk] = G[(size_t)(ntile * 16 + rr) * N_NODES + kk];
  }
  __syncthreads();

  int wavein = threadIdx.x >> 5;
  int lane = threadIdx.x & 31;
  int r16 = lane & 15, hi = lane >> 4;
  int tiles_c = C >> 4;
  int cid = blockIdx.y * 8 + wavein;           // over BATCH * tiles_c
  int b = cid / tiles_c, tc = cid % tiles_c;
  int col = tc * 16 + r16;
  const float* xb = x + (size_t)b * N_NODES * C;
  const float* gsrow = Gs + r16 * GSTRIDE + 2 * hi;
  v8f acc = {};
  for (int k0 = 0; k0 < N_NODES; k0 += 4) {
    v2f av = {gsrow[k0], gsrow[k0 + 1]};       // A[m=r16][k0+2hi +{0,1}]
    int ka = k0 + 2 * hi;
    v2f bv = {xb[(size_t)ka * C + col], xb[(size_t)(ka + 1) * C + col]};
    acc = __builtin_amdgcn_wmma_f32_16x16x4_f32(false, av, false, bv, (short)0, acc,
                                                false, false);
  }
  float* o = out + ((size_t)b * N_NODES + ntile * 16 + 8 * hi) * C + col;
#pragma unroll
  for (int r = 0; r < 8; ++r) o[(size_t)r * C] = acc[r];
}

// ---------------- BatchNorm stats: mean/rstd per channel over ROWS ----------------

__global__ __launch_bounds__(256) void k_stats(const float* __restrict__ src, int C,
                                               float* mean, float* rstd) {
  int c = blockIdx.x;
  double s = 0.0, s2 = 0.0;
  for (int r = threadIdx.x; r < ROWS; r += blockDim.x) {
    float v = src[(size_t)r * C + c];
    s += v; s2 += (double)v * v;
  }
  __shared__ double sh[256], sh2[256];
  sh[threadIdx.x] = s; sh2[threadIdx.x] = s2;
  __syncthreads();
  for (int o = 128; o > 0; o >>= 1) {
    if (threadIdx.x < o) { sh[threadIdx.x] += sh[threadIdx.x + o];
                           sh2[threadIdx.x] += sh2[threadIdx.x + o]; }
    __syncthreads();
  }
  if (threadIdx.x == 0) {
    double m = sh[0] / ROWS;
    double var = sh2[0] / ROWS - m * m;
    mean[c] = (float)m;
    rstd[c] = rsqrtf((float)var + 1e-5f);
  }
}

// ---------------- gate 1: conc = sigmoid(bn_relu(g1)+x1+b1); rh = r*h (flat split) ----

__global__ void k_gate1(const float* __restrict__ g1, const float* __restrict__ x1,
                        const float* __restrict__ mean, const float* __restrict__ rstd,
                        const float* __restrict__ gamma, const float* __restrict__ beta,
                        const float* __restrict__ bias, const float* __restrict__ h,
                        float* __restrict__ conc, float* __restrict__ rh) {
  int idx = blockIdx.x * blockDim.x + threadIdx.x;  // BATCH * 65536
  int b = idx >> 16, rem = idx & 65535, c = rem & 127;
  float v = (g1[idx] - mean[c]) * rstd[c] * gamma[c] + beta[c];
  v = fmaxf(v, 0.f) + x1[idx] + bias[c];
  float sgm = 1.f / (1.f + expf(-v));
  conc[idx] = sgm;
  if (rem < NH) rh[(size_t)b * NH + rem] = sgm * h[(size_t)b * NH + rem];
}

// ---------------- gate 2: c = tanh(bn_relu(g2)+x2+b2); h = u*h+(1-u)*c; emit feat ----

__global__ void k_gate2(const float* __restrict__ g2, const float* __restrict__ x2,
                        const float* __restrict__ mean, const float* __restrict__ rstd,
                        const float* __restrict__ gamma, const float* __restrict__ beta,
                        const float* __restrict__ bias, const float* __restrict__ conc,
                        float* __restrict__ h, float* __restrict__ feat_base,
                        size_t bstride, size_t toff) {
  int idx = blockIdx.x * blockDim.x + threadIdx.x;  // BATCH * NH
  int b = idx >> 15, i = idx & 32767, c = i & 63;
  float v = (g2[idx] - mean[c]) * rstd[c] * gamma[c] + beta[c];
  v = fmaxf(v, 0.f) + x2[idx] + bias[c];
  float cc = tanhf(v);
  float u = conc[(size_t)b * 65536 + NH + i];       // flat-index split: u = 2nd half
  float hn = u * h[idx] + (1.f - u) * cc;
  h[idx] = hn;
  feat_base[toff + (size_t)b * bstride + i] = hn;
}

// ---------------- output layer (tiny: C_out = 2) ----------------

__global__ void k_outmm(const float* __restrict__ o0, const float* __restrict__ f1,
                        const float* __restrict__ W0, const float* __restrict__ W1,
                        float* __restrict__ h0p, float* __restrict__ h1p) {
  int idx = blockIdx.x * blockDim.x + threadIdx.x;  // BATCH*N*2
  int c = idx & 1, bn = idx >> 1;
  const float* a0 = o0 + (size_t)bn * HD;
  const float* a1 = f1 + (size_t)bn * HD;
  float s0 = 0.f, s1 = 0.f;
  for (int k = 0; k < HD; ++k) { s0 += a0[k] * W0[k * 2 + c]; s1 += a1[k] * W1[k * 2 + c]; }
  h0p[idx] = s0; h1p[idx] = s1;
}

__global__ void k_outgconv(const float* __restrict__ G, const float* __restrict__ hp,
                           float* __restrict__ gout) {
  int idx = blockIdx.x * blockDim.x + threadIdx.x;  // BATCH*N*2
  int c = idx & 1, bn = idx >> 1, b = bn >> 9, n = bn & 511;
  const float* gr = G + (size_t)n * N_NODES;
  const float* hb = hp + (size_t)b * N_NODES * 2 + c;
  float s = 0.f;
  for (int m = 0; m < N_NODES; ++m) s += gr[m] * hb[m * 2];
  gout[idx] = s;
}

__global__ void k_outfinal(const float* __restrict__ g0, const float* __restrict__ g1,
                           const float* __restrict__ h0p, const float* __restrict__ h1p,
                           const float* m0, const float* r0,
                           const float* m1, const float* r1,
                           const float* gamma, const float* beta, const float* bias,
                           int t, float* __restrict__ pred_out) {
  int idx = blockIdx.x * blockDim.x + threadIdx.x;  // BATCH*N*2
  int c = idx & 1, bn = idx >> 1, b = bn >> 9, n = bn & 511;
  float v0 = fmaxf((g0[idx] - m0[c]) * r0[c] * gamma[c] + beta[c], 0.f) + h0p[idx];
  float v1 = fmaxf((g1[idx] - m1[c]) * r1[c] * gamma[c] + beta[c], 0.f) + h1p[idx];
  float p = 0.5f * (v0 + v1) + bias[c];
  pred_out[((size_t)b * SEQ + t) * N_NODES * 2 + (size_t)n * 2 + c] = p;
}

// ---------------- host orchestration ----------------

extern "C" void kernel_launch(void* const* d_in, const int* in_sizes, int n_in,
                              void* d_out, int out_size, void* d_ws, size_t ws_size,
                              hipStream_t stream) {
  const float* inputs    = (const float*)d_in[0];
  const float* inc       = (const float*)d_in[1];
  const float* gc1_W     = (const float*)d_in[2];
  const float* gc1_b     = (const float*)d_in[3];
  const float* gc1_gamma = (const float*)d_in[4];
  const float* gc1_beta  = (const float*)d_in[5];
  const float* gc2_W     = (const float*)d_in[6];
  const float* gc2_b     = (const float*)d_in[7];
  const float* gc2_gamma = (const float*)d_in[8];
  const float* gc2_beta  = (const float*)d_in[9];
  const float* out_W0    = (const float*)d_in[10];
  const float* out_W1    = (const float*)d_in[11];
  const float* out_b     = (const float*)d_in[12];
  const float* out_gamma = (const float*)d_in[13];
  const float* out_beta  = (const float*)d_in[14];

  float* preds  = (float*)d_out;                         // (B,S,N,2)
  float* feats0 = (float*)d_out + (size_t)BATCH * SEQ * N_NODES * 2;  // (B,S,N,64)

  float* w = (float*)d_ws;
  float* G     = w; w += (size_t)N_NODES * N_NODES;
  float* dv    = w; w += 512;
  float* de    = w; w += 128;
  float* h     = w; w += (size_t)BATCH * NH;
  float* x1    = w; w += (size_t)BATCH * N_NODES * 128;
  float* g1b   = w; w += (size_t)BATCH * N_NODES * 128;
  float* conc  = w; w += (size_t)BATCH * N_NODES * 128;
  float* rh    = w; w += (size_t)BATCH * NH;
  float* x2    = w; w += (size_t)BATCH * NH;
  float* g2b   = w; w += (size_t)BATCH * NH;
  float* h0p   = w; w += (size_t)BATCH * N_NODES * 2;
  float* h1p   = w; w += (size_t)BATCH * N_NODES * 2;
  float* g0o   = w; w += (size_t)BATCH * N_NODES * 2;
  float* g1o   = w; w += (size_t)BATCH * N_NODES * 2;
  float* mean1 = w; w += 128;
  float* rstd1 = w; w += 128;
  float* mean2 = w; w += 64;
  float* rstd2 = w; w += 64;
  float* m0    = w; w += 64;
  float* r0    = w; w += 64;
  float* m1    = w; w += 64;
  float* r1    = w; w += 64;
  float* feats1 = w;  // (S,B,NH) = 33,554,432 floats

  // graph operator
  k_deg<<<1, 512, 0, stream>>>(inc, dv, de);
  k_buildG<<<N_NODES, 512, 0, stream>>>(inc, dv, de, G);

  auto cell = [&](int t) {
    k_catmm<<<1024, 256, 0, stream>>>(inputs, t, h, gc1_W, 128, x1);
    k_gconv<<<dim3(32, 32), 256, 0, stream>>>(G, x1, 128, g1b);
    k_stats<<<128, 256, 0, stream>>>(g1b, 128, mean1, rstd1);
    k_gate1<<<(BATCH * 65536) / 256, 256, 0, stream>>>(g1b, x1, mean1, rstd1,
        gc1_gamma, gc1_beta, gc1_b, h, conc, rh);
    k_catmm<<<512, 256, 0, stream>>>(inputs, t, rh, gc2_W, 64, x2);
    k_gconv<<<dim3(32, 16), 256, 0, stream>>>(G, x2, 64, g2b);
    k_stats<<<64, 256, 0, stream>>>(g2b, 64, mean2, rstd2);
  };

  // backward scan: h=0; t = S-1 .. 0; feats1[t] = h
  k_fill<<<(BATCH * NH) / 256, 256, 0, stream>>>(h, 0.f, BATCH * NH);
  for (int t = SEQ - 1; t >= 0; --t) {
    cell(t);
    k_gate2<<<(BATCH * NH) / 256, 256, 0, stream>>>(g2b, x2, mean2, rstd2,
        gc2_gamma, gc2_beta, gc2_b, conc, h,
        feats1, (size_t)NH, (size_t)t * BATCH * NH);
  }

  // forward scan: h=0; t = 0 .. S-1; feats0[b][t] = h; pred via outlayer(h, feats1[t])
  k_fill<<<(BATCH * NH) / 256, 256, 0, stream>>>(h, 0.f, BATCH * NH);
  for (int t = 0; t < SEQ; ++t) {
    cell(t);
    k_gate2<<<(BATCH * NH) / 256, 256, 0, stream>>>(g2b, x2, mean2, rstd2,
        gc2_gamma, gc2_beta, gc2_b, conc, h,
        feats0, (size_t)SEQ * NH, (size_t)t * NH);
    const float* f1 = feats1 + (size_t)t * BATCH * NH;
    k_outmm<<<128, 256, 0, stream>>>(h, f1, out_W0, out_W1, h0p, h1p);
    k_outgconv<<<128, 256, 0, stream>>>(G, h0p, g0o);
    k_outgconv<<<128, 256, 0, stream>>>(G, h1p, g1o);
    k_stats<<<2, 256, 0, stream>>>(g0o, 2, m0, r0);
    k_stats<<<2, 256, 0, stream>>>(g1o, 2, m1, r1);
    k_outfinal<<<128, 256, 0, stream>>>(g0o, g1o, h0p, h1p, m0, r0, m1, r1,
        out_gamma, out_beta, out_b, t, preds);
  }
}